// LocSE_90640989815381
// MI455X (gfx1250) — compile-verified
//
#include <hip/hip_runtime.h>
#include <hip/hip_bf16.h>

// CDNA5 / gfx1250. Wave32. WMMA f32 16x16x4 for the tiny MLP GEMM.

typedef __attribute__((ext_vector_type(2))) float v2f;
typedef __attribute__((ext_vector_type(8))) float v8f;

#define NBLK 512
#define BLK  256
#define KTOP 16

// ---------------------------------------------------------------------------
// Pass 1: streaming top-16 selection. Memory-bound: one pass over 96 MB
// (~4.1 us at 23.3 TB/s). Two rows per thread -> b128+b64+b32 per pair
// instead of 6 scalar b32 loads. Each thread keeps a sorted top-16 of
// (dist2_bits<<32 | index) keys in registers (steady state: 1 compare per
// point). Block reduces 256x16 candidates in LDS to a sorted top-16 -> d_ws.
// ---------------------------------------------------------------------------
__global__ __launch_bounds__(BLK) void locse_topk_partial(
    const float* __restrict__ xf, const int* __restrict__ idxp,
    unsigned long long* __restrict__ cand, int N)
{
    const int cidx = idxp[0];
    const float cx = xf[cidx * 6 + 0];
    const float cy = xf[cidx * 6 + 1];
    const float cz = xf[cidx * 6 + 2];

    unsigned long long best[KTOP];
#pragma unroll
    for (int t = 0; t < KTOP; ++t) best[t] = ~0ull;

    const int tid    = threadIdx.x;
    const int gid    = blockIdx.x * BLK + tid;
    const int stride = NBLK * BLK;
    const int npair  = N >> 1;

    for (int p = gid; p < npair; p += stride) {
        if (p + stride < npair)
            __builtin_prefetch(xf + (size_t)(p + stride) * 12, 0, 0);  // global_prefetch_b8

        // row pair 2p, 2p+1 occupies floats [12p .. 12p+11]; 16B-aligned base
        const float4 v0  = *(const float4*)(xf + (size_t)p * 12);      // x0 y0 z0 (f3)
        const float2 v1  = *(const float2*)(xf + (size_t)p * 12 + 6);  // x1 y1
        const float  z1  = xf[(size_t)p * 12 + 8];

        {
            const float dx = v0.x - cx, dy = v0.y - cy, dz = v0.z - cz;
            const float d2 = dx * dx + dy * dy + dz * dz;
            const unsigned long long key =
                ((unsigned long long)__float_as_uint(d2) << 32) | (unsigned int)(2 * p);
            if (key < best[KTOP - 1]) {
                int j = KTOP - 1;
                while (j > 0 && best[j - 1] > key) { best[j] = best[j - 1]; --j; }
                best[j] = key;
            }
        }
        {
            const float dx = v1.x - cx, dy = v1.y - cy, dz = z1 - cz;
            const float d2 = dx * dx + dy * dy + dz * dz;
            const unsigned long long key =
                ((unsigned long long)__float_as_uint(d2) << 32) | (unsigned int)(2 * p + 1);
            if (key < best[KTOP - 1]) {
                int j = KTOP - 1;
                while (j > 0 && best[j - 1] > key) { best[j] = best[j - 1]; --j; }
                best[j] = key;
            }
        }
    }

    // odd tail row (N even in practice; handled for generality)
    if ((N & 1) && gid == 0) {
        const int i = N - 1;
        const float dx = xf[i * 6 + 0] - cx;
        const float dy = xf[i * 6 + 1] - cy;
        const float dz = xf[i * 6 + 2] - cz;
        const float d2 = dx * dx + dy * dy + dz * dz;
        const unsigned long long key =
            ((unsigned long long)__float_as_uint(d2) << 32) | (unsigned int)i;
        if (key < best[KTOP - 1]) {
            int j = KTOP - 1;
            while (j > 0 && best[j - 1] > key) { best[j] = best[j - 1]; --j; }
            best[j] = key;
        }
    }

    __shared__ unsigned long long lds[BLK * KTOP];   // 32 KB of the 320 KB WGP LDS
    __shared__ unsigned long long rmin[BLK];
#pragma unroll
    for (int t = 0; t < KTOP; ++t) lds[tid * KTOP + t] = best[t];
    __syncthreads();

    for (int r = 0; r < KTOP; ++r) {
        unsigned long long m = lds[tid * KTOP];
        int pos = 0;
#pragma unroll
        for (int t = 1; t < KTOP; ++t) {
            unsigned long long v = lds[tid * KTOP + t];
            if (v < m) { m = v; pos = t; }
        }
        rmin[tid] = m;
        __syncthreads();
        for (int s = BLK / 2; s > 0; s >>= 1) {
            if (tid < s) {
                unsigned long long o = rmin[tid + s];
                if (o < rmin[tid]) rmin[tid] = o;
            }
            __syncthreads();
        }
        const unsigned long long gm = rmin[0];
        __syncthreads();
        if (m == gm)                        // keys are globally unique
            lds[tid * KTOP + pos] = ~0ull;  // consume winner
        if (tid == 0) cand[blockIdx.x * KTOP + r] = gm;
        __syncthreads();
    }
}

// ---------------------------------------------------------------------------
// Pass 2: single wave32. Select global top-16 from 512x16 candidates, then
// run the 16x10 @ 10x3 MLP as 3 chained V_WMMA_F32_16X16X4_F32 (K=10 -> 12),
// bias preloaded into the f32 accumulator. Writes the 16x6 output.
// A-fragment values are built with one select per component (lanes 0-15 vs
// 16-31); B-fragment weights come from LDS via a clamped branchless index.
// ---------------------------------------------------------------------------
__global__ __launch_bounds__(32) void locse_finalize(
    const float* __restrict__ xf, const float* __restrict__ W,
    const float* __restrict__ bvec, const int* __restrict__ idxp,
    const unsigned long long* __restrict__ cand, float* __restrict__ out)
{
    __shared__ unsigned long long keys[NBLK * KTOP];  // 64 KB
    __shared__ unsigned long long rmin[32];
    __shared__ int   nn[KTOP];
    __shared__ float Wsh[32];                         // 30 weights, padded
    const int lane = threadIdx.x;  // 0..31, full wave

    if (lane < 30) Wsh[lane] = W[lane];
    for (int i = lane; i < NBLK * KTOP; i += 32) keys[i] = cand[i];
    __syncthreads();

    for (int r = 0; r < KTOP; ++r) {
        unsigned long long m = ~0ull; int pos = -1;
        for (int i = lane; i < NBLK * KTOP; i += 32) {
            unsigned long long v = keys[i];
            if (v < m) { m = v; pos = i; }
        }
        rmin[lane] = m;
        __syncthreads();
        if (lane == 0) {
            unsigned long long g = rmin[0];
            for (int t = 1; t < 32; ++t) if (rmin[t] < g) g = rmin[t];
            rmin[0] = g;
        }
        __syncthreads();
        const unsigned long long gm = rmin[0];
        if (pos >= 0 && m == gm) {          // unique winner marks its slot
            keys[pos] = ~0ull;
            nn[r] = (int)(unsigned int)(gm & 0xffffffffull);
        }
        __syncthreads();
    }

    // ---- MLP: r = mlp_in(16x10) @ W^T(10x3) + b, fp32 WMMA -----------------
    const int cidx = idxp[0];
    const float cx = xf[cidx * 6 + 0], cy = xf[cidx * 6 + 1], cz = xf[cidx * 6 + 2];

    // NOTE reference quirk: nn_pts = xyz[:16] (first 16 rows), not neighbors.
    const int M = lane & 15;                      // A row index
    const float px = xf[M * 6 + 0], py = xf[M * 6 + 1], pz = xf[M * 6 + 2];
    const float dx = px - cx, dy = py - cy, dz = pz - cz;
    const float dn = sqrtf(dx * dx + dy * dy + dz * dz);
    // mlp_in columns: {cx,cy,cz, px,py,pz, dx,dy,dz, dn, 0, 0}

    const int  Ncol = lane & 15;                  // N index for B/C/D layout
    const bool hi   = lane >= 16;

    // C/D layout: VGPR v <-> M=v (lanes 0-15) / M=v+8 (lanes 16-31); N = lane&15.
    const float cval = (Ncol < 3) ? bvec[Ncol] : 0.f;
    v8f c = {cval, cval, cval, cval, cval, cval, cval, cval};

    // A 16x4 f32 fragments: lanes 0-15 -> {K=4ch, 4ch+1}; lanes 16-31 -> {4ch+2, 4ch+3}
    const v2f a0 = { hi ? cz  : cx, hi ? px  : cy };   // K 0,1 | 2,3
    const v2f a1 = { hi ? dx  : py, hi ? dy  : pz };   // K 4,5 | 6,7
    const v2f a2 = { hi ? 0.f : dz, hi ? 0.f : dn };   // K 8,9 | 10,11 (pad)

    // B 4x16 f32 fragments: VGPR v -> K = 4ch+v (lo) / 4ch+v+2 (hi); N = lane&15.
    // B[k][n] = W[n][k]; rows n>=3 and k>=10 are zero.
    const int   wrow  = (Ncol < 3) ? Ncol : 0;         // clamped row (branchless)
    const float nmask = (Ncol < 3) ? 1.f : 0.f;
    const int   kb    = hi ? 2 : 0;
    const float m2    = hi ? 0.f : nmask;              // chunk 2 hi lanes -> K=10,11 pad
    const v2f b0 = { Wsh[wrow * 10 + kb + 0] * nmask, Wsh[wrow * 10 + kb + 1] * nmask };
    const v2f b1 = { Wsh[wrow * 10 + kb + 4] * nmask, Wsh[wrow * 10 + kb + 5] * nmask };
    const v2f b2 = { Wsh[wrow * 10 + 8] * m2,         Wsh[wrow * 10 + 9] * m2 };

    c = __builtin_amdgcn_wmma_f32_16x16x4_f32(false, a0, false, b0, (short)0, c, false, false);
    c = __builtin_amdgcn_wmma_f32_16x16x4_f32(false, a1, false, b1, (short)0, c, false, false);
    c = __builtin_amdgcn_wmma_f32_16x16x4_f32(false, a2, false, b2, (short)0, c, false, false);

    // out row m: [r(m,0..2), f(m,0..2)], out is 16x6
    if (Ncol < 3) {
        const int mbase = hi ? 8 : 0;
#pragma unroll
        for (int v = 0; v < 8; ++v)
            out[(mbase + v) * 6 + Ncol] = c[v];
    }
    if (lane < 16) {
        const int id = nn[lane];
        out[lane * 6 + 3] = xf[id * 6 + 0];
        out[lane * 6 + 4] = xf[id * 6 + 1];
        out[lane * 6 + 5] = xf[id * 6 + 2];
    }
}

// ---------------------------------------------------------------------------
extern "C" void kernel_launch(void* const* d_in, const int* in_sizes, int n_in,
                              void* d_out, int out_size, void* d_ws, size_t ws_size,
                              hipStream_t stream)
{
    const float* xf   = (const float*)d_in[0];   // (N,6) f32
    const float* W    = (const float*)d_in[1];   // (3,10) f32
    const float* bv   = (const float*)d_in[2];   // (3,)  f32
    const int*   idxp = (const int*)d_in[3];     // scalar idx
    float* out        = (float*)d_out;           // (16,6) f32

    const int N = in_sizes[0] / 6;
    unsigned long long* cand = (unsigned long long*)d_ws;  // 512*16*8 = 64 KB

    locse_topk_partial<<<NBLK, BLK, 0, stream>>>(xf, idxp, cand, N);
    locse_finalize<<<1, 32, 0, stream>>>(xf, W, bv, idxp, cand, out);
}